// GaussianKDE_11467562680978
// MI455X (gfx1250) — compile-verified
//
#include <hip/hip_runtime.h>

typedef __attribute__((ext_vector_type(2))) float v2f;
typedef __attribute__((ext_vector_type(8))) float v8f;

#define DCOLS   64
#define ABLOCKS 250
#define RED_OFF 16384   // float offset into workspace for reduced stats

// ---------------------------------------------------------------------------
// Kernel 1: per-block partial reduction of dataset.
// Produces partials[b*65 + c] = sum over block-rows of dataset[:, c]  (c<64)
//          partials[b*65 + 64] = sum over block-rows of ||d_row||^2
// Thread t handles column t&63, row-phase t>>6 (4 rows per iteration,
// fully coalesced 64-lane row reads).
// ---------------------------------------------------------------------------
__global__ __launch_bounds__(256)
void kde_reduce_dataset(const float* __restrict__ dataset,
                        float* __restrict__ partials,
                        int N, int rowsPerBlock) {
    __shared__ float s1[256];
    __shared__ float s2[256];
    const int t    = threadIdx.x;
    const int col  = t & 63;
    const int rq   = t >> 6;
    const int base = blockIdx.x * rowsPerBlock;

    float acc = 0.0f, accsq = 0.0f;
    for (int r = rq; r < rowsPerBlock; r += 4) {
        const int row = base + r;
        if (row < N) {
            const float v = dataset[row * DCOLS + col];
            acc   += v;
            accsq += v * v;
        }
    }
    s1[t] = acc;
    s2[t] = accsq;
    __syncthreads();
    if (t < 64) {
        const float cs = s1[t] + s1[t + 64] + s1[t + 128] + s1[t + 192];
        partials[blockIdx.x * 65 + t] = cs;
    }
    if (t == 0) {
        float sq = 0.0f;
        for (int i = 0; i < 256; ++i) sq += s2[i];
        partials[blockIdx.x * 65 + 64] = sq;
    }
}

// ---------------------------------------------------------------------------
// Kernel 2: fold 250 partials -> reduced[0..63] = dbar (mean dataset row),
//           reduced[64] = log_norm - 0.5 * mean(||d||^2).  Deterministic.
// ---------------------------------------------------------------------------
__global__ __launch_bounds__(128)
void kde_finalize_stats(const float* __restrict__ partials,
                        const float* __restrict__ log_norm,
                        float* __restrict__ reduced,
                        int N) {
    const int t = threadIdx.x;
    if (t < 64) {
        float s = 0.0f;
        for (int b = 0; b < ABLOCKS; ++b) s += partials[b * 65 + t];
        reduced[t] = s / (float)N;
    } else if (t == 64) {
        float s = 0.0f;
        for (int b = 0; b < ABLOCKS; ++b) s += partials[b * 65 + 64];
        reduced[64] = log_norm[0] - 0.5f * (s / (float)N);
    }
}

// ---------------------------------------------------------------------------
// Kernel 3: one wave32 per 16 query rows.
// f_tile[16,64] = features_tile[16,64] @ bw[64,64] via v_wmma_f32_16x16x4_f32
// (4 N-tiles x 16 K-steps = 64 WMMAs), then fused epilogue:
//   out_i = reduced[64] + sum_c ( f[i][c]*dbar[c] - 0.5*f[i][c]^2 )
// A frag (16x4, 32-bit): lane m=L&15, VGPRs = K = {2*(L>>4), 2*(L>>4)+1}
// B frag (4x16, 32-bit): lane n=L&15, VGPRs = same K pair
// D (16x16, 32-bit):     VGPR r, lane L -> M = r + 8*(L>>4), N = L&15
// EXEC is all-ones for the whole WMMA region (no divergence before it).
// ---------------------------------------------------------------------------
__global__ __launch_bounds__(32)
void kde_wmma_eval(const float* __restrict__ features,
                   const float* __restrict__ bw,
                   const float* __restrict__ reduced,
                   float* __restrict__ out) {
    __shared__ float red[256];
    const int lane = threadIdx.x;          // 0..31
    const int half = lane >> 4;            // 0 or 1
    const int n16  = lane & 15;            // 0..15
    const int k0   = 2 * half;             // K offset within a 4-wide K step
    const int q0   = blockIdx.x * 16;

    const float* arow = features + (q0 + n16) * DCOLS;

    float tacc[8] = {0.f, 0.f, 0.f, 0.f, 0.f, 0.f, 0.f, 0.f};

#pragma unroll
    for (int nt = 0; nt < 4; ++nt) {
        v8f acc = {0.f, 0.f, 0.f, 0.f, 0.f, 0.f, 0.f, 0.f};
#pragma unroll
        for (int kk = 0; kk < 16; ++kk) {
            const int k = 4 * kk + k0;
            v2f a = *(const v2f*)(arow + k);          // A[m][k], A[m][k+1]
            v2f b;
            b.x = bw[(k    ) * DCOLS + nt * 16 + n16]; // B[k  ][n]
            b.y = bw[(k + 1) * DCOLS + nt * 16 + n16]; // B[k+1][n]
            acc = __builtin_amdgcn_wmma_f32_16x16x4_f32(
                false, a, false, b, (short)0, acc, false, false);
        }
        const float db = reduced[nt * 16 + n16];       // dbar[col]
#pragma unroll
        for (int r = 0; r < 8; ++r) {
            const float fv = acc[r];                   // f[r+8*half][16*nt+n16]
            tacc[r] += fv * db - 0.5f * fv * fv;
        }
    }

#pragma unroll
    for (int r = 0; r < 8; ++r)
        red[(r + 8 * half) * 16 + n16] = tacc[r];
    __syncthreads();

    if (lane < 16) {
        float s = reduced[64];                         // log_norm - 0.5*mean(d2)
        for (int j = 0; j < 16; ++j) s += red[lane * 16 + j];
        out[q0 + lane] = s;
    }
}

// ---------------------------------------------------------------------------
extern "C" void kernel_launch(void* const* d_in, const int* in_sizes, int n_in,
                              void* d_out, int out_size, void* d_ws, size_t ws_size,
                              hipStream_t stream) {
    const float* features = (const float*)d_in[0];   // [Q, 64]
    const float* bw       = (const float*)d_in[1];   // [64, 64]
    const float* dataset  = (const float*)d_in[2];   // [N, 64]
    const float* log_norm = (const float*)d_in[3];   // [1]
    float* out = (float*)d_out;                      // [Q]

    float* wsf      = (float*)d_ws;
    float* partials = wsf;                 // ABLOCKS * 65 floats
    float* reduced  = wsf + RED_OFF;       // 65 floats

    const int N = in_sizes[2] / DCOLS;
    const int rowsPerBlock = (N + ABLOCKS - 1) / ABLOCKS;

    kde_reduce_dataset<<<ABLOCKS, 256, 0, stream>>>(dataset, partials, N, rowsPerBlock);
    kde_finalize_stats<<<1, 128, 0, stream>>>(partials, log_norm, reduced, N);

    const int qtiles = out_size / 16;      // Q = 2048 -> 128 tiles
    kde_wmma_eval<<<qtiles, 32, 0, stream>>>(features, bw, reduced, out);
}